// StandardMHA_42047729828043
// MI455X (gfx1250) — compile-verified
//
#include <hip/hip_runtime.h>

#define EMBED   1024
#define NHEADS  16
#define HD      64
#define SEQ     2048
#define BATCH   2
#define SCALE_F 0.125f      // HEAD_DIM^-0.5
#define NEG_F   (-1.0e9f)

// gfx1250 async global->LDS copy (ASYNCcnt-tracked), guarded so the kernel
// still compiles (with the manual staging path) if the builtins are absent.
#if defined(__has_builtin)
#  if __has_builtin(__builtin_amdgcn_global_load_async_to_lds_b128) && \
      __has_builtin(__builtin_amdgcn_s_wait_asynccnt)
#    define USE_ASYNC_LDS 1
#  endif
#endif
#ifndef USE_ASYNC_LDS
#  define USE_ASYNC_LDS 0
#endif

typedef float v2f __attribute__((ext_vector_type(2)));
typedef float v8f __attribute__((ext_vector_type(8)));
typedef int   v4i __attribute__((ext_vector_type(4)));

#define GLOBAL_AS __attribute__((address_space(1)))
#define LDS_AS    __attribute__((address_space(3)))

static __device__ __forceinline__ v8f wmma_f32(v2f a, v2f b, v8f c) {
    // D = A(16x4 f32) * B(4x16 f32) + C(16x16 f32)
    return __builtin_amdgcn_wmma_f32_16x16x4_f32(false, a, false, b,
                                                 (short)0, c, false, false);
}

// ---------------------------------------------------------------------------
// Projection GEMM: Y[r][c] = sum_k X[r][k] * W[c][k] + bias[c]
// X: [BATCH*SEQ, EMBED] row-major, W: [EMBED, EMBED] row-major (torch Linear).
// headSplit=1: write Y in [B, H, S, HD] layout; headSplit=0: [B*S, EMBED].
// Block: 128 threads = 4 waves; each wave computes a 32x64 output tile
// (2 row-tiles x 4 col-tiles), so each K-step is 6 b64 loads per 8 WMMAs.
// ---------------------------------------------------------------------------
__global__ __launch_bounds__(128, 1)
void proj_kernel(const float* __restrict__ X, const float* __restrict__ W,
                 const float* __restrict__ bias, float* __restrict__ Y,
                 int headSplit)
{
    const int lane = threadIdx.x & 31;
    const int wave = threadIdx.x >> 5;
    const int l16  = lane & 15;
    const int half = lane >> 4;

    const int row0 = blockIdx.x * 128 + wave * 32;  // global row in [0, B*S)
    const int col0 = blockIdx.y * 64;               // output column base

    // fp32 WMMA lane mapping: lane holds (row l16, k = half*2 + {0,1}) pairs
    const float* xrow0 = X + (size_t)(row0 + l16) * EMBED + half * 2;
    const float* xrow1 = xrow0 + (size_t)16 * EMBED;
    const float* wrow  = W + (size_t)(col0 + l16) * EMBED + half * 2;

    v8f a00 = {}, a01 = {}, a02 = {}, a03 = {};
    v8f a10 = {}, a11 = {}, a12 = {}, a13 = {};

    #pragma unroll 4
    for (int k = 0; k < EMBED; k += 4) {
        v2f a0 = *(const v2f*)(xrow0 + k);
        v2f a1 = *(const v2f*)(xrow1 + k);
        v2f b0 = *(const v2f*)(wrow + k);
        v2f b1 = *(const v2f*)(wrow + 16 * EMBED + k);
        v2f b2 = *(const v2f*)(wrow + 32 * EMBED + k);
        v2f b3 = *(const v2f*)(wrow + 48 * EMBED + k);
        a00 = wmma_f32(a0, b0, a00);  a10 = wmma_f32(a1, b0, a10);
        a01 = wmma_f32(a0, b1, a01);  a11 = wmma_f32(a1, b1, a11);
        a02 = wmma_f32(a0, b2, a02);  a12 = wmma_f32(a1, b2, a12);
        a03 = wmma_f32(a0, b3, a03);  a13 = wmma_f32(a1, b3, a13);
    }

    #pragma unroll
    for (int rt = 0; rt < 2; ++rt) {
        #pragma unroll
        for (int t = 0; t < 4; ++t) {
            v8f acc = (rt == 0)
                ? ((t == 0) ? a00 : (t == 1) ? a01 : (t == 2) ? a02 : a03)
                : ((t == 0) ? a10 : (t == 1) ? a11 : (t == 2) ? a12 : a13);
            const int c = col0 + t * 16 + l16;
            const float bb = bias[c];
            #pragma unroll
            for (int v = 0; v < 8; ++v) {
                const int m = rt * 16 + v + 8 * half;   // row within 32-row tile
                const int r = row0 + m;
                const float val = acc[v] + bb;
                if (headSplit) {
                    const int b = r / SEQ, s = r % SEQ;
                    const int h = col0 >> 6;            // 64 cols per head
                    const int d = t * 16 + l16;
                    Y[(((size_t)(b * NHEADS + h) * SEQ) + s) * HD + d] = val;
                } else {
                    Y[(size_t)r * EMBED + c] = val;
                }
            }
        }
    }
}

// ---------------------------------------------------------------------------
// Causal flash attention. Block = 128 threads = 4 waves; the block owns 4
// consecutive 16-row Q tiles of one (b,h) and the waves share each 16-key
// K/V tile staged in LDS. K tile goes through the gfx1250 async global->LDS
// path when available; V is transposed into LDS manually so the P*V
// B-operand is a single conflict-free ds_load_b64.
// Q,K,V: [B, H, S, HD].  O: [B, S, EMBED] (head h at columns h*64..h*64+63).
// ---------------------------------------------------------------------------
__global__ __launch_bounds__(128, 1)
void attn_kernel(const float* __restrict__ Q, const float* __restrict__ K,
                 const float* __restrict__ V, float* __restrict__ O)
{
    __shared__ __align__(16) float klds[16 * 68];      // K tile, row stride 68
    __shared__ __align__(16) float vtld[64 * 18];      // V^T tile, row stride 18
    __shared__ __align__(16) float plds[4][16 * 18];   // per-wave P tile

    const int tid  = threadIdx.x;
    const int lane = tid & 31;
    const int wave = tid >> 5;
    const int l16  = lane & 15;
    const int half = lane >> 4;

    const int bh    = blockIdx.y;                 // b*NHEADS + h
    const int qt    = blockIdx.x * 4 + wave;      // this wave's q tile
    const int qtmax = blockIdx.x * 4 + 3;         // block-uniform loop bound
    const int q0    = qt * 16;

    const float* Qb = Q + (size_t)bh * SEQ * HD;
    const float* Kb = K + (size_t)bh * SEQ * HD;
    const float* Vb = V + (size_t)bh * SEQ * HD;

    // Preload Q tile into fp32-WMMA A-layout registers: qa[st] covers k=4*st..+3
    v2f qa[16];
    {
        const float* qrow = Qb + (size_t)(q0 + l16) * HD + half * 2;
        #pragma unroll
        for (int st = 0; st < 16; ++st)
            qa[st] = *(const v2f*)(qrow + 4 * st);
    }

    v8f acc0 = {}, acc1 = {}, acc2 = {}, acc3 = {};
    float mrow[8], srow[8];
    #pragma unroll
    for (int v = 0; v < 8; ++v) { mrow[v] = -3.0e38f; srow[v] = 0.0f; }

    float* pl = plds[wave];

    // cooperative-staging coordinates: 128 threads x 8 floats = 16x64 tile
    const int sr = tid >> 3;         // 0..15  (key row within tile)
    const int sc = (tid & 7) * 8;    // 0,8,..,56 (dim column)

    for (int kb = 0; kb <= qtmax; ++kb) {
        const int k0 = kb * 16;

        // ---- stage K tile (row-major, pad 68) ----
#if USE_ASYNC_LDS
        {
            const float* kg = Kb + (size_t)(k0 + sr) * HD + sc;
            float* kl = klds + sr * 68 + sc;
            __builtin_amdgcn_global_load_async_to_lds_b128(
                (GLOBAL_AS v4i*)kg, (LDS_AS v4i*)kl, 0, 0);
            __builtin_amdgcn_global_load_async_to_lds_b128(
                (GLOBAL_AS v4i*)kg, (LDS_AS v4i*)kl, 16, 0);
        }
#else
        {
            const float4* kg = (const float4*)(Kb + (size_t)(k0 + sr) * HD + sc);
            float4 ka = kg[0], kc = kg[1];
            *(float4*)(klds + sr * 68 + sc)     = ka;
            *(float4*)(klds + sr * 68 + sc + 4) = kc;
        }
#endif
        // ---- stage V tile transposed (pad 18) ----
        {
            const float4* vg = (const float4*)(Vb + (size_t)(k0 + sr) * HD + sc);
            float4 va = vg[0], vc = vg[1];
            vtld[(sc + 0) * 18 + sr] = va.x;
            vtld[(sc + 1) * 18 + sr] = va.y;
            vtld[(sc + 2) * 18 + sr] = va.z;
            vtld[(sc + 3) * 18 + sr] = va.w;
            vtld[(sc + 4) * 18 + sr] = vc.x;
            vtld[(sc + 5) * 18 + sr] = vc.y;
            vtld[(sc + 6) * 18 + sr] = vc.z;
            vtld[(sc + 7) * 18 + sr] = vc.w;
        }
#if USE_ASYNC_LDS
        __builtin_amdgcn_s_wait_asynccnt(0);   // barrier does not cover ASYNCcnt
#endif
        __syncthreads();

        if (kb <= qt) {
            const bool diag = (kb == qt);

            // ---- S = Q * K^T (contraction over HD=64, 16 WMMAs) ----
            v8f s = {};
            #pragma unroll
            for (int st = 0; st < 16; ++st) {
                v2f b = *(const v2f*)(klds + l16 * 68 + half * 2 + 4 * st);
                s = wmma_f32(qa[st], b, s);
            }

            // ---- scale + causal mask ----
            #pragma unroll
            for (int v = 0; v < 8; ++v) {
                float val = s[v] * SCALE_F;
                if (diag) {
                    const int m = v + 8 * half;
                    if (k0 + l16 > q0 + m) val = NEG_F;
                }
                s[v] = val;
            }

            // ---- online softmax (row stats per 16-lane half-group) ----
            #pragma unroll
            for (int v = 0; v < 8; ++v) {
                float tmax = s[v];
                tmax = fmaxf(tmax, __shfl_xor(tmax, 1, 16));
                tmax = fmaxf(tmax, __shfl_xor(tmax, 2, 16));
                tmax = fmaxf(tmax, __shfl_xor(tmax, 4, 16));
                tmax = fmaxf(tmax, __shfl_xor(tmax, 8, 16));
                const float mnew = fmaxf(mrow[v], tmax);
                const float corr = __expf(mrow[v] - mnew);
                mrow[v] = mnew;
                const float p = __expf(s[v] - mnew);
                s[v] = p;
                float psum = p;
                psum += __shfl_xor(psum, 1, 16);
                psum += __shfl_xor(psum, 2, 16);
                psum += __shfl_xor(psum, 4, 16);
                psum += __shfl_xor(psum, 8, 16);
                srow[v] = srow[v] * corr + psum;
                acc0[v] *= corr; acc1[v] *= corr; acc2[v] *= corr; acc3[v] *= corr;
            }

            // ---- P: D-layout -> LDS -> A-layout (same wave, in order) ----
            #pragma unroll
            for (int v = 0; v < 8; ++v)
                pl[(v + 8 * half) * 18 + l16] = s[v];

            // ---- acc += P * V (16 keys; 4 col tiles x 4 K-steps) ----
            #pragma unroll
            for (int st = 0; st < 4; ++st) {
                v2f pa = *(const v2f*)(pl + l16 * 18 + half * 2 + 4 * st);
                const int kk = half * 2 + 4 * st;
                v2f b0 = *(const v2f*)(vtld + ( 0 + l16) * 18 + kk);
                v2f b1 = *(const v2f*)(vtld + (16 + l16) * 18 + kk);
                v2f b2 = *(const v2f*)(vtld + (32 + l16) * 18 + kk);
                v2f b3 = *(const v2f*)(vtld + (48 + l16) * 18 + kk);
                acc0 = wmma_f32(pa, b0, acc0);
                acc1 = wmma_f32(pa, b1, acc1);
                acc2 = wmma_f32(pa, b2, acc2);
                acc3 = wmma_f32(pa, b3, acc3);
            }
        }
        __syncthreads();   // protect K/V LDS tiles before next stage
    }

    // ---- normalize and write out in [B, S, EMBED] layout ----
    const int b = bh / NHEADS, h = bh % NHEADS;
    #pragma unroll
    for (int v = 0; v < 8; ++v) {
        const float inv = 1.0f / srow[v];
        const int m = v + 8 * half;
        const size_t rowoff = ((size_t)b * SEQ + (q0 + m)) * EMBED + h * HD;
        O[rowoff +      l16] = acc0[v] * inv;
        O[rowoff + 16 + l16] = acc1[v] * inv;
        O[rowoff + 32 + l16] = acc2[v] * inv;
        O[rowoff + 48 + l16] = acc3[v] * inv;
    }
}

// ---------------------------------------------------------------------------
extern "C" void kernel_launch(void* const* d_in, const int* in_sizes, int n_in,
                              void* d_out, int out_size, void* d_ws, size_t ws_size,
                              hipStream_t stream)
{
    (void)in_sizes; (void)n_in; (void)out_size; (void)ws_size;
    const float* x  = (const float*)d_in[0];
    // d_in[1] = attn_mask (bool causal) -- causality is hardcoded
    const float* Wq = (const float*)d_in[2];
    const float* bq = (const float*)d_in[3];
    const float* Wk = (const float*)d_in[4];
    const float* bk = (const float*)d_in[5];
    const float* Wv = (const float*)d_in[6];
    const float* bv = (const float*)d_in[7];
    const float* Wo = (const float*)d_in[8];
    const float* bo = (const float*)d_in[9];

    float* out  = (float*)d_out;
    float* Kout = out  + (size_t)BATCH * SEQ * EMBED;                // K in return order
    float* Vout = Kout + (size_t)BATCH * NHEADS * SEQ * HD;          // V after K

    float* Qws     = (float*)d_ws;                                   // [B,H,S,HD]
    float* attn_ws = Qws + (size_t)BATCH * SEQ * EMBED;              // [B,S,EMBED]

    const dim3 blk(128);
    const dim3 gproj(BATCH * SEQ / 128, EMBED / 64);

    proj_kernel<<<gproj, blk, 0, stream>>>(x, Wq, bq, Qws,  1);
    proj_kernel<<<gproj, blk, 0, stream>>>(x, Wk, bk, Kout, 1);
    proj_kernel<<<gproj, blk, 0, stream>>>(x, Wv, bv, Vout, 1);
    attn_kernel<<<dim3(SEQ / 64, BATCH * NHEADS), blk, 0, stream>>>(Qws, Kout, Vout, attn_ws);
    proj_kernel<<<gproj, blk, 0, stream>>>(attn_ws, Wo, bo, out, 0);
}